// TemporalReceptiveField_31284541784737
// MI455X (gfx1250) — compile-verified
//
#include <hip/hip_runtime.h>

// ---------------------------------------------------------------------------
// LIF / SuperSpike forward step, (B,C,H,W) = (16,64,128,128), fp32.
//   v_decayed = v0 + DT*tau_c*(i0 - v0)
//   i_decayed = (1 - DT*TAU_SYN_INV) * i0          (= 0.8*i0)
//   z         = (v_decayed - V_TH > 0) ? 1 : 0
//   v_new     = z ? V_RESET(=0) : v_decayed
//   i_new     = i_decayed + x
//
// Pure streaming kernel: 402 MB of traffic, ~17 us floor @ 23.3 TB/s.
// CDNA5 path: b128 VMEM, TH=NT load/store (data never fits 192MB L2 and is
// touched once), SGPR-uniform per-channel tau via 2-D grid (plane = b*C+c).
// ---------------------------------------------------------------------------

typedef __attribute__((ext_vector_type(4))) float v4f;

namespace {
constexpr float kDT        = 0.001f;
constexpr float kTauSynInv = 200.0f;
constexpr float kVth       = 1.0f;

constexpr int kC          = 64;
constexpr int kHW         = 128 * 128;   // floats per (b,c) plane = 16384
constexpr int kV4PerPlane = kHW / 4;     // 4096 float4 units per plane
constexpr int kThreads    = 256;         // 8 waves (wave32)
constexpr int kUnroll     = 2;           // 2 float4 per thread
constexpr int kBlocksX    = kV4PerPlane / (kThreads * kUnroll);  // 8, exact
}

__global__ __launch_bounds__(kThreads)
void lif_step_kernel(const float* __restrict__ x,
                     const float* __restrict__ v0,
                     const float* __restrict__ i0,
                     const float* __restrict__ ps,
                     float* __restrict__ out,   // [ z | v_new | i_new ]
                     long n)                    // elements per array
{
    const int  plane = blockIdx.y;                     // b*C + c  (uniform)
    const float tau_dt = kDT * ps[plane & (kC - 1)];   // SGPR load + broadcast
    const float keep   = 1.0f - kDT * kTauSynInv;      // 0.8

    const long planeBase = (long)plane * kHW;
    float* __restrict__ zOut = out;
    float* __restrict__ vOut = out + n;
    float* __restrict__ iOut = out + 2 * n;

#pragma unroll
    for (int u = 0; u < kUnroll; ++u) {
        const int  q    = (blockIdx.x * kUnroll + u) * kThreads + threadIdx.x;
        const long base = planeBase + (long)q * 4;

        // Streaming reads: touched exactly once -> non-temporal b128 loads.
        const v4f xr = __builtin_nontemporal_load((const v4f*)(x  + base));
        const v4f vr = __builtin_nontemporal_load((const v4f*)(v0 + base));
        const v4f ir = __builtin_nontemporal_load((const v4f*)(i0 + base));

        // v_decayed = v0 + tau_dt*(i0 - v0)   (one sub + one fma per lane-elem)
        const v4f vd   = vr + tau_dt * (ir - vr);
        // i_new = 0.8*i0 + x                  (one fma per lane-elem)
        const v4f iNew = keep * ir + xr;

        v4f z, vNew;
#pragma unroll
        for (int k = 0; k < 4; ++k) {
            const bool spike = vd[k] > kVth;      // v_cmp_gt_f32
            z[k]    = spike ? 1.0f : 0.0f;        // v_cndmask
            vNew[k] = spike ? 0.0f : vd[k];       // v_cndmask
        }

        // Streaming writes: never re-read -> non-temporal b128 stores,
        // bypass/rinse L2 instead of polluting 192MB of it.
        __builtin_nontemporal_store(z,    (v4f*)(zOut + base));
        __builtin_nontemporal_store(vNew, (v4f*)(vOut + base));
        __builtin_nontemporal_store(iNew, (v4f*)(iOut + base));
    }
}

extern "C" void kernel_launch(void* const* d_in, const int* in_sizes, int n_in,
                              void* d_out, int out_size, void* d_ws, size_t ws_size,
                              hipStream_t stream) {
    const float* x  = (const float*)d_in[0];
    const float* v0 = (const float*)d_in[1];
    const float* i0 = (const float*)d_in[2];
    const float* ps = (const float*)d_in[3];
    float* out = (float*)d_out;

    const long n      = (long)in_sizes[0];        // 16,777,216 elements
    const int  planes = (int)(n / kHW);           // B*C = 1024

    dim3 grid(kBlocksX, planes, 1);               // (8, 1024)
    lif_step_kernel<<<grid, kThreads, 0, stream>>>(x, v0, i0, ps, out, n);
}